// AEVComputer_33054068310393
// MI455X (gfx1250) — compile-verified
//
#include <hip/hip_runtime.h>

typedef float v2f __attribute__((ext_vector_type(2)));
typedef float v8f __attribute__((ext_vector_type(8)));

#define A_ATOMS 48
#define NPAIRS  1128          // 48*47/2, and 1128 == 282*4 exactly
#define NGROUPS (NPAIRS / 4)  // 282 WMMA K=4 steps
#define RCRf 5.2f
#define RCAf 3.5f
#define PI_F 3.14159265358979323846f

__global__ __launch_bounds__(128) void aev_kernel(
    const float* __restrict__ coords,   // (M,48,3)
    const float* __restrict__ EtaR,     // (1)
    const float* __restrict__ ShfR,     // (16)
    const float* __restrict__ EtaA,     // (1)
    const float* __restrict__ Zeta,     // (1)
    const float* __restrict__ ShfA,     // (4)
    const float* __restrict__ ShfZ,     // (8)
    const int*   __restrict__ species,  // (M,48)
    const int*   __restrict__ triu,     // (4,4) (unused: arithmetic triu index)
    float*       __restrict__ out)      // (M,48,384)
{
    __shared__ float  sC[A_ATOMS][3];
    __shared__ float4 sP1[A_ATOMS];   // {vx, vy, vz, d}
    __shared__ float4 sP2[A_ATOMS];   // {1/d, fca*wa, species(bitcast), 0}
    __shared__ int    sSP[A_ATOMS];
    __shared__ unsigned short sPair[NPAIRS];  // j | (k<<8)
    __shared__ float  bins[384];      // [0,64) radial s*16+r ; [64,384) angular p*32+c

    const int blk = blockIdx.x;
    const int m   = blk / A_ATOMS;
    const int i   = blk % A_ATOMS;
    const int tid = threadIdx.x;

    for (int t = tid; t < A_ATOMS * 3; t += 128)
        sC[t / 3][t % 3] = coords[(size_t)m * A_ATOMS * 3 + t];
    for (int t = tid; t < A_ATOMS; t += 128)
        sSP[t] = species[m * A_ATOMS + t];
    for (int t = tid; t < 384; t += 128) bins[t] = 0.0f;
    // build (j<k) pair list
    for (int t = tid; t < NPAIRS; t += 128) {
        int j = 0, rem = t;
        while (rem >= (A_ATOMS - 1 - j)) { rem -= (A_ATOMS - 1 - j); ++j; }
        sPair[t] = (unsigned short)(j | ((j + 1 + rem) << 8));
    }
    __syncthreads();

    const float ci0 = sC[i][0], ci1 = sC[i][1], ci2 = sC[i][2];
    const int   spi = sSP[i];
    const float etaR = EtaR[0], etaA = EtaA[0], zeta = Zeta[0];

    // per-neighbor packed state
    for (int j = tid; j < A_ATOMS; j += 128) {
        float vx = sC[j][0] - ci0, vy = sC[j][1] - ci1, vz = sC[j][2] - ci2;
        float d2 = vx * vx + vy * vy + vz * vz;
        float d  = sqrtf(d2 > 0.0f ? d2 : 1.0f);
        float fca = 0.5f * cosf(d * (PI_F / RCAf)) + 0.5f;
        bool  w   = (spi >= 0) && (sSP[j] >= 0) && (j != i) && (d <= RCAf);
        int   spc = sSP[j] >= 0 ? sSP[j] : 0;
        sP1[j] = make_float4(vx, vy, vz, d);
        sP2[j] = make_float4(__builtin_amdgcn_rcpf(d), w ? fca : 0.0f,
                             __int_as_float(spc), 0.0f);
    }
    __syncthreads();

    // ---------------- radial part (VALU + LDS atomics) ----------------
    if (tid < A_ATOMS) {
        const int j = tid;
        const float d = sP1[j].w;
        const bool mr = (spi >= 0) && (sSP[j] >= 0) && (j != i) && (d <= RCRf);
        if (mr) {
            const float fc = 0.5f * cosf(d * (PI_F / RCRf)) + 0.5f;
            const int s = sSP[j];
            #pragma unroll
            for (int r = 0; r < 16; ++r) {
                float df  = d - ShfR[r];
                float val = 0.25f * __expf(-etaR * df * df) * fc;
                atomicAdd(&bins[s * 16 + r], val);
            }
        }
    }

    // ---------------- angular part (WMMA f32 16x16x4) ----------------
    // Per wave: K=4 jk-pairs per step. A[M=channel, K=pair], B[K=pair, N=pairclass(one-hot)]
    // A layout (32-bit, 16x4): lanes 0-15 -> K=0(V0),1(V1); lanes 16-31 -> K=2(V0),3(V1)
    const int wave = tid >> 5;
    const int lane = tid & 31;
    const int half = lane >> 4;
    const int n    = lane & 15;                 // channel row (A) / pair-class col (B)
    const float shfz    = ShfZ[n & 7];
    const float cz = cosf(shfz), sz = sinf(shfz);
    const float shfA_lo = ShfA[n >> 3];         // channels 0..15  : sa = n>>3
    const float shfA_hi = ShfA[(n >> 3) + 2];   // channels 16..31 : sa = (n>>3)+2

    v8f c0 = {}; // channels 0..15  x pair-classes
    v8f c1 = {}; // channels 16..31 x pair-classes

    for (int g = wave; g < NGROUPS; g += 4) {
        const int pa = g * 4 + half * 2;   // this half-wave owns pairs pa, pa+1

        float tLo[2], tHi[2]; int pc[2];
        #pragma unroll
        for (int q = 0; q < 2; ++q) {
            const unsigned pp = sPair[pa + q];
            const int jp = pp & 0xFF, kp = pp >> 8;
            const float4 aj = sP1[jp], ak = sP1[kp];
            const float4 bj = sP2[jp], bk = sP2[kp];
            const float dot = aj.x * ak.x + aj.y * ak.y + aj.z * ak.z;
            float cosv = 0.95f * dot * bj.x * bk.x;
            cosv = fminf(0.99f, fmaxf(-0.99f, cosv));
            // cos(acos(c) - shfz) = c*cos(shfz) + sqrt(1-c^2)*sin(shfz)
            const float sinv   = sqrtf(1.0f - cosv * cosv);
            const float coseff = cosv * cz + sinv * sz;
            const float base   = 0.5f + 0.5f * coseff;
            const float f1     = __powf(base, zeta);     // same sz for lo & hi channels
            const float davg   = 0.5f * (aj.w + ak.w);
            const float fcpm   = 2.0f * bj.y * bk.y;     // fca*wa products, mask folded in
            const float dlo = davg - shfA_lo;
            const float dhi = davg - shfA_hi;
            const float f1m = f1 * fcpm;
            tLo[q] = f1m * __expf(-etaA * dlo * dlo);
            tHi[q] = f1m * __expf(-etaA * dhi * dhi);
            const int sj = __float_as_int(bj.z);
            const int sk = __float_as_int(bk.z);
            const int smin = sj < sk ? sj : sk;
            const int smax = sj < sk ? sk : sj;
            pc[q] = (smin * (9 - smin)) / 2 + (smax - smin); // np.triu_indices(4) order
        }

        v2f a0, a1, b;
        a0.x = tLo[0]; a0.y = tLo[1];
        a1.x = tHi[0]; a1.y = tHi[1];
        b.x = (n == pc[0]) ? 1.0f : 0.0f;   // one-hot pair-class column
        b.y = (n == pc[1]) ? 1.0f : 0.0f;

        c0 = __builtin_amdgcn_wmma_f32_16x16x4_f32(false, a0, false, b,
                                                   (short)0, c0, false, false);
        c1 = __builtin_amdgcn_wmma_f32_16x16x4_f32(false, a1, false, b,
                                                   (short)0, c1, false, false);
    }

    // D layout: lane<16 -> D[M=v, N=lane]; lane>=16 -> D[M=v+8, N=lane-16]
    if (n < 10) {
        const int Mbase = half * 8;
        #pragma unroll
        for (int v = 0; v < 8; ++v) {
            atomicAdd(&bins[64 + n * 32 + (Mbase + v)],      c0[v]);
            atomicAdd(&bins[64 + n * 32 + (16 + Mbase + v)], c1[v]);
        }
    }
    __syncthreads();

    float* outp = out + (size_t)(m * A_ATOMS + i) * 384;
    for (int t = tid; t < 384; t += 128) outp[t] = bins[t];
}

extern "C" void kernel_launch(void* const* d_in, const int* in_sizes, int n_in,
                              void* d_out, int out_size, void* d_ws, size_t ws_size,
                              hipStream_t stream) {
    const float* coords  = (const float*)d_in[0];
    const float* etaR    = (const float*)d_in[1];
    const float* shfR    = (const float*)d_in[2];
    const float* etaA    = (const float*)d_in[3];
    const float* zeta    = (const float*)d_in[4];
    const float* shfA    = (const float*)d_in[5];
    const float* shfZ    = (const float*)d_in[6];
    const int*   species = (const int*)d_in[7];
    const int*   triu    = (const int*)d_in[8];
    float*       out     = (float*)d_out;

    const int M = in_sizes[7] / A_ATOMS;   // species is (M,48)
    aev_kernel<<<dim3(M * A_ATOMS), dim3(128), 0, stream>>>(
        coords, etaR, shfR, etaA, zeta, shfA, shfZ, species, triu, out);
}